// attention_net_16819091931353
// MI455X (gfx1250) — compile-verified
//
#include <hip/hip_runtime.h>
#include <math.h>

#define M_SIZE      8192
#define FEATURE_DIM 256
#define EMBED_DIM   128
#define ALPHA       0.2f
#define MASK_VAL    (-9000000000000000.0f)

typedef float v2f __attribute__((ext_vector_type(2)));
typedef float v8f __attribute__((ext_vector_type(8)));

// ---------------------------------------------------------------------------
// Kernel A: w1[t] = sum_e W[t][e]*a[e];  w2[t] = sum_e W[t][e]*a[128+e]
// W is FEATURE_DIM x EMBED_DIM row-major. Trivial cost (64K MACs).
// ---------------------------------------------------------------------------
__global__ void project_a(const float* __restrict__ W,
                          const float* __restrict__ a,
                          float* __restrict__ w12) {
    int t = threadIdx.x;            // 0..255, one thread per feature row
    float acc1 = 0.f, acc2 = 0.f;
    #pragma unroll 4
    for (int e = 0; e < EMBED_DIM; ++e) {
        float wv = W[t * EMBED_DIM + e];
        acc1 = fmaf(wv, a[e], acc1);
        acc2 = fmaf(wv, a[EMBED_DIM + e], acc2);
    }
    w12[t]               = acc1;
    w12[FEATURE_DIM + t] = acc2;
}

// ---------------------------------------------------------------------------
// Kernel B: s = V @ w via V_WMMA_F32_16X16X4_F32 (full f32 precision).
// One wave (32 lanes) per 16-row tile. B is the w-chunk broadcast into all
// 16 columns, so every column of D equals the GEMV result for this tile.
//
// A 16x4 f32 layout (ISA 7.12.2): lanes 0-15 -> M=lane, VGPR0=K+0, VGPR1=K+1;
//                                 lanes 16-31 -> M=lane-16, VGPR0=K+2, VGPR1=K+3.
// B 4x16 layout mirrored: VGPR0 = row K+0 (lanes 0-15) / row K+2 (lanes 16-31),
//                         VGPR1 = row K+1 / row K+3. Broadcast columns make
//                         only the K->slot mapping matter.
// D (C/D 16x16 f32): VGPR g: lanes 0-15 = M=g, lanes 16-31 = M=8+g.
// ---------------------------------------------------------------------------
__global__ void gemv_wmma(const float* __restrict__ v_i,
                          const float* __restrict__ v_j,
                          const float* __restrict__ w12,
                          float* __restrict__ s_all) {
    const int  wave  = blockIdx.x;               // 0..1023, blockDim.x == 32
    const int  ntile = M_SIZE / 16;              // 512 tiles per matrix
    const bool is_j  = wave >= ntile;
    const int  tile  = is_j ? (wave - ntile) : wave;

    const float* __restrict__ V = is_j ? v_j : v_i;
    const float* __restrict__ w = w12 + (is_j ? FEATURE_DIM : 0);
    float*       __restrict__ s = s_all + (is_j ? M_SIZE : 0);

    const int lane = threadIdx.x;                // 0..31
    const int m    = lane & 15;
    const int hi   = lane >> 4;                  // 0: K+0/K+1, 1: K+2/K+3

    const float* vrow = V + (size_t)(tile * 16 + m) * FEATURE_DIM;

    v8f c = {};                                  // accumulator = 0
    #pragma unroll 4
    for (int k = 0; k < FEATURE_DIM; k += 4) {
        const int kk = k + hi * 2;
        v2f aop, bop;
        aop.x = vrow[kk];
        aop.y = vrow[kk + 1];
        bop.x = w[kk];
        bop.y = w[kk + 1];
        // (neg_a, A, neg_b, B, c_mod, C, reuse_a, reuse_b)
        c = __builtin_amdgcn_wmma_f32_16x16x4_f32(false, aop, false, bop,
                                                  (short)0, c, false, false);
    }

    // All columns of D are identical; lane 0 holds rows 0..7, lane 16 rows 8..15.
    if (m == 0) {
        #pragma unroll
        for (int g = 0; g < 8; ++g)
            s[tile * 16 + hi * 8 + g] = c[g];
    }
}

// ---------------------------------------------------------------------------
// Kernel C: per-row masked leaky-relu softmax. One 1024-thread block per row
// (32 waves). 8 elements/thread kept in registers -> adj is read exactly once
// (non-temporal), output written once (non-temporal). s_j (32KB) stays hot in
// cache. HBM traffic ~512MB total => ~22us roofline at 23.3 TB/s.
// ---------------------------------------------------------------------------
#define ROW_THREADS 1024
#define PER_THREAD  (M_SIZE / ROW_THREADS)       // 8

__global__ void __launch_bounds__(ROW_THREADS)
masked_softmax(const float* __restrict__ s_all,
               const int*   __restrict__ adj,
               float*       __restrict__ out) {
    const int row = blockIdx.x;
    const int tid = threadIdx.x;

    const float* __restrict__ s_j = s_all + M_SIZE;
    const float  si = s_all[row];

    const int*   __restrict__ arow = adj + (size_t)row * M_SIZE;
    float*       __restrict__ orow = out + (size_t)row * M_SIZE;

    float e[PER_THREAD];
    float lmax = MASK_VAL;

    #pragma unroll
    for (int c = 0; c < PER_THREAD; ++c) {
        const int j  = tid + c * ROW_THREADS;    // coalesced 4KB per step
        const int mv = __builtin_nontemporal_load(arow + j);
        const float sc = si + s_j[j];
        const float lr = fmaxf(sc, ALPHA * sc);  // leaky_relu, ALPHA < 1
        const float ev = (mv > 0) ? lr : MASK_VAL;
        e[c] = ev;
        lmax = fmaxf(lmax, ev);
    }

    // ---- block max reduction (wave32 shuffle + LDS across 32 waves) ----
    __shared__ float red[32];
    #pragma unroll
    for (int off = 16; off > 0; off >>= 1)
        lmax = fmaxf(lmax, __shfl_xor(lmax, off, 32));
    const int wid = tid >> 5;
    if ((tid & 31) == 0) red[wid] = lmax;
    __syncthreads();
    if (wid == 0) {
        float v = red[tid & 31];
        #pragma unroll
        for (int off = 16; off > 0; off >>= 1)
            v = fmaxf(v, __shfl_xor(v, off, 32));
        red[tid & 31] = v;                       // broadcast slot 0..31 all equal
    }
    __syncthreads();
    const float rmax = red[0];

    // ---- exp + block sum reduction ----
    float lsum = 0.f;
    #pragma unroll
    for (int c = 0; c < PER_THREAD; ++c) {
        const float p = __expf(e[c] - rmax);     // masked entries -> exp(-huge) = 0
        e[c] = p;
        lsum += p;
    }
    __syncthreads();                              // red[] reuse
    #pragma unroll
    for (int off = 16; off > 0; off >>= 1)
        lsum += __shfl_xor(lsum, off, 32);
    if ((tid & 31) == 0) red[wid] = lsum;
    __syncthreads();
    if (wid == 0) {
        float v = red[tid & 31];
        #pragma unroll
        for (int off = 16; off > 0; off >>= 1)
            v += __shfl_xor(v, off, 32);
        red[tid & 31] = v;
    }
    __syncthreads();
    const float inv = 1.0f / red[0];

    #pragma unroll
    for (int c = 0; c < PER_THREAD; ++c) {
        const int j = tid + c * ROW_THREADS;
        __builtin_nontemporal_store(e[c] * inv, orow + j);
    }
}

// ---------------------------------------------------------------------------
extern "C" void kernel_launch(void* const* d_in, const int* in_sizes, int n_in,
                              void* d_out, int out_size, void* d_ws, size_t ws_size,
                              hipStream_t stream) {
    const float* v_i = (const float*)d_in[0];
    const float* v_j = (const float*)d_in[1];
    const int*   adj = (const int*)  d_in[2];
    const float* W   = (const float*)d_in[3];
    const float* a   = (const float*)d_in[4];
    float* out = (float*)d_out;

    float* ws    = (float*)d_ws;
    float* w12   = ws;                 // 2 * FEATURE_DIM floats
    float* s_all = ws + 2 * FEATURE_DIM; // 2 * M_SIZE floats (s_i then s_j)

    project_a<<<1, FEATURE_DIM, 0, stream>>>(W, a, w12);
    gemv_wmma<<<2 * (M_SIZE / 16), 32, 0, stream>>>(v_i, v_j, w12, s_all);
    masked_softmax<<<M_SIZE, ROW_THREADS, 0, stream>>>(s_all, adj, out);
}